// conformer_1133871366342
// MI455X (gfx1250) — compile-verified
//
#include <hip/hip_runtime.h>
#include <hip/hip_bf16.h>
#include <math.h>

typedef __attribute__((ext_vector_type(2))) float v2f;
typedef __attribute__((ext_vector_type(8))) float v8f;

#define DEPTH 3
#define EMB 40
#define HEADS 8
#define HDIM 5
#define BATCH 256
#define SEQ 7
#define FEAT 280
#define OUTN 512
#define T1 176          // 200 - 25 + 1
#define H62 62
#define KTAP 25
#define HROWS 64        // 62 h-rows padded to 64 per tap segment
#define KPAD (KTAP * HROWS)   // 1600
#define NPAD 48         // oc 40 padded to 3x16
#define POOLW 75
#define POOLS 15
#define FC_K 280
#define AS_STR 284      // LDS stride (mult of 4 for b128; bank-clean)
#define TTILE 32        // t positions per workgroup (two 16-row M tiles)
#define XW 56           // TTILE + 24 halo

// ---------------- Kernel A0: fold conv1*conv2*BNscale into one filter ----------------
// Layout: Wc2[kt][h][ocp], kt-major segments of 64 h-rows (rows 62,63 zero), N padded to 48.
__global__ void fold_weights(const float* __restrict__ w1,   // (40,1,1,25) [ic][k]
                             const float* __restrict__ w2,   // (40,40,62,1) [oc][ic][h]
                             const float* __restrict__ gamma,
                             const float* __restrict__ var,
                             float* __restrict__ Wc2)        // (KPAD, NPAD) K-major
{
    int idx = blockIdx.x * blockDim.x + threadIdx.x;
    if (idx >= KPAD * NPAD) return;
    int kk  = idx / NPAD;
    int ocp = idx - kk * NPAD;
    int kt  = kk >> 6;          // tap
    int h   = kk & 63;          // spatial row
    float val = 0.f;
    if (h < H62 && ocp < EMB) {
        float acc = 0.f;
        for (int ic = 0; ic < 40; ++ic)
            acc += w2[(ocp * 40 + ic) * 62 + h] * w1[ic * 25 + kt];
        float scale = gamma[ocp] * rsqrtf(var[ocp] + 1e-5f);
        val = acc * scale;
    }
    Wc2[kk * NPAD + ocp] = val;
}

// ---------------- Kernel A1: fold biases + BN shift ----------------
__global__ void fold_bias(const float* __restrict__ b1, const float* __restrict__ w2,
                          const float* __restrict__ b2, const float* __restrict__ gamma,
                          const float* __restrict__ beta, const float* __restrict__ mean,
                          const float* __restrict__ var, float* __restrict__ biasC)
{
    int oc = threadIdx.x;
    if (oc >= EMB) return;
    float acc = b2[oc];
    for (int ic = 0; ic < 40; ++ic) {
        float s = 0.f;
        for (int h = 0; h < 62; ++h) s += w2[(oc * 40 + ic) * 62 + h];
        acc += s * b1[ic];
    }
    float scale = gamma[oc] * rsqrtf(var[oc] + 1e-5f);
    biasC[oc] = (acc - mean[oc]) * scale + beta[oc];
}

// ---------------- Kernel B: fused conv as implicit GEMM (f32 WMMA) + ELU ----------------
// grid (6 t-tiles of 32, 256 batch), block 96 (3 waves; wave w -> oc tile w*16)
// Each wave: 32x16 output via two accumulators sharing every B fragment.
__global__ __launch_bounds__(96) void conv_wmma_kernel(
    const float* __restrict__ x,     // (256,1,62,200)
    const float* __restrict__ Wc2,   // (KPAD, NPAD)
    const float* __restrict__ biasC, // (40)
    float* __restrict__ eluOut)      // (256,40,176)
{
    __shared__ float xs[HROWS * XW];   // 64 rows x 56 (rows 62,63 zeroed)
    const int t0  = blockIdx.x * TTILE;
    const int b   = blockIdx.y;
    const int tid = threadIdx.x;

    const float* xb = x + (size_t)b * (H62 * 200);
    if (t0 + XW <= 200) {
        // interior tile: two-phase float4 staging (868 vec4 = 62 rows x 14)
        float4 v[10];
        #pragma unroll
        for (int j = 0; j < 10; ++j) {
            int i = tid + j * 96;
            if (i < H62 * (XW / 4)) {
                int h = i / (XW / 4), c = i - h * (XW / 4);
                v[j] = *(const float4*)(xb + h * 200 + t0 + c * 4);
            }
        }
        #pragma unroll
        for (int j = 0; j < 10; ++j) {
            int i = tid + j * 96;
            if (i < H62 * (XW / 4)) {
                int h = i / (XW / 4), c = i - h * (XW / 4);
                *(float4*)&xs[h * XW + c * 4] = v[j];
            }
        }
    } else {
        // last tile: clamped scalar staging (clamped cols only feed unwritten rows)
        for (int i = tid; i < H62 * XW; i += 96) {
            int h = i / XW, tt = i - h * XW;
            int c = t0 + tt; c = c > 199 ? 199 : c;
            xs[h * XW + tt] = xb[h * 200 + c];
        }
    }
    // zero the two pad rows so 0-weight rows never touch NaN garbage
    for (int i = tid; i < 2 * XW; i += 96) xs[H62 * XW + i] = 0.f;
    __syncthreads();

    const int lane  = tid & 31;
    const int wave  = tid >> 5;
    const int noff  = wave * 16;
    const int m     = lane & 15;          // M for A, N for B
    const int khalf = (lane >> 4) << 1;   // 0 or 2 (upper half-wave K rows)

    v8f acc0 = {};
    v8f acc1 = {};
    for (int kt = 0; kt < KTAP; ++kt) {
        const float* wseg = Wc2 + (size_t)kt * HROWS * NPAD + noff + m;
        const float* xrow = xs + m + kt;
        #pragma unroll 4
        for (int h4 = 0; h4 < HROWS; h4 += 4) {
            const int r0 = h4 + khalf;
            float a00 = xrow[r0 * XW];
            float a01 = xrow[(r0 + 1) * XW];
            float a10 = xrow[r0 * XW + 16];
            float a11 = xrow[(r0 + 1) * XW + 16];
            v2f Bv = { wseg[r0 * NPAD], wseg[(r0 + 1) * NPAD] };
            v2f A0 = { a00, a01 };
            v2f A1 = { a10, a11 };
            acc0 = __builtin_amdgcn_wmma_f32_16x16x4_f32(false, A0, false, Bv,
                                                         (short)0, acc0, false, false);
            acc1 = __builtin_amdgcn_wmma_f32_16x16x4_f32(false, A1, false, Bv,
                                                         (short)0, acc1, false, false);
        }
    }

    const int rbase = (lane >> 4) * 8;
    const int oc    = noff + m;
    if (oc < EMB) {
        float bia = biasC[oc];
        float* outp = eluOut + ((size_t)b * EMB + oc) * T1;
        #pragma unroll
        for (int r = 0; r < 8; ++r) {
            int t = t0 + rbase + r;                 // always < 176
            float z = acc0[r] + bia;
            outp[t] = z > 0.f ? z : expm1f(z);      // ELU(alpha=1)
        }
        #pragma unroll
        for (int r = 0; r < 8; ++r) {
            int t = t0 + 16 + rbase + r;
            if (t < T1) {
                float z = acc1[r] + bia;
                outp[t] = z > 0.f ? z : expm1f(z);
            }
        }
    }
}

// ---------------- Kernel C: average pool 75/15 ----------------
__global__ void pool_kernel(const float* __restrict__ eluOut, float* __restrict__ pooled)
{
    int idx = blockIdx.x * blockDim.x + threadIdx.x;
    if (idx >= BATCH * EMB * SEQ) return;
    int tp = idx % SEQ;
    int rest = idx / SEQ;
    int oc = rest % EMB;
    int b  = rest / EMB;
    const float* p = eluOut + ((size_t)b * EMB + oc) * T1 + tp * POOLS;
    float s = 0.f;
    for (int j = 0; j < POOLW; ++j) s += p[j];
    pooled[idx] = s * (1.f / 75.f);
}

// ---------------- Kernel D: 1x1 proj + 3 encoder blocks, one WG per batch elem ----------------
__global__ __launch_bounds__(256) void transformer_kernel(
    const float* __restrict__ pooled,  // (256,40,7)
    const float* __restrict__ projw, const float* __restrict__ projb,
    const float* __restrict__ ln1g, const float* __restrict__ ln1b,
    const float* __restrict__ wq, const float* __restrict__ bq,
    const float* __restrict__ wk, const float* __restrict__ bk,
    const float* __restrict__ wv, const float* __restrict__ bv,
    const float* __restrict__ wo, const float* __restrict__ bo,
    const float* __restrict__ ln2g, const float* __restrict__ ln2b,
    const float* __restrict__ fw1, const float* __restrict__ fb1,
    const float* __restrict__ fw2, const float* __restrict__ fb2,
    float* __restrict__ flat)          // (256,280)
{
    __shared__ float s[SEQ][EMB];
    __shared__ float tn[SEQ][EMB];
    __shared__ float qm[SEQ][EMB];
    __shared__ float km[SEQ][EMB];
    __shared__ float vm[SEQ][EMB];
    __shared__ float ao[SEQ][EMB];
    __shared__ float hid[SEQ][4 * EMB];
    __shared__ float mu[SEQ], rsd[SEQ];
    const int b = blockIdx.x, tid = threadIdx.x;

    for (int i = tid; i < SEQ * EMB; i += 256) {
        int t = i / EMB, e = i - t * EMB;
        float a = projb[e];
        for (int ic = 0; ic < EMB; ++ic)
            a += projw[e * EMB + ic] * pooled[(b * EMB + ic) * SEQ + t];
        s[t][e] = a;
    }
    __syncthreads();
    const float sc_att = rsqrtf((float)EMB);   // reference scales by 1/sqrt(EMB)

    for (int L = 0; L < DEPTH; ++L) {
        if (tid < SEQ) {
            float m = 0.f;
            for (int e = 0; e < EMB; ++e) m += s[tid][e];
            m *= (1.f / EMB);
            float v = 0.f;
            for (int e = 0; e < EMB; ++e) { float d = s[tid][e] - m; v += d * d; }
            mu[tid] = m; rsd[tid] = rsqrtf(v * (1.f / EMB) + 1e-5f);
        }
        __syncthreads();
        for (int i = tid; i < SEQ * EMB; i += 256) {
            int t = i / EMB, e = i - t * EMB;
            tn[t][e] = (s[t][e] - mu[t]) * rsd[t] * ln1g[L * EMB + e] + ln1b[L * EMB + e];
        }
        __syncthreads();
        for (int i = tid; i < 3 * SEQ * EMB; i += 256) {
            int which = i / (SEQ * EMB);
            int r = i - which * SEQ * EMB;
            int t = r / EMB, e = r - t * EMB;
            const float* W  = (which == 0 ? wq : which == 1 ? wk : wv) + (size_t)L * EMB * EMB + e * EMB;
            const float* Bv = (which == 0 ? bq : which == 1 ? bk : bv) + L * EMB;
            float a = Bv[e];
            for (int j = 0; j < EMB; ++j) a += W[j] * tn[t][j];
            (which == 0 ? qm : which == 1 ? km : vm)[t][e] = a;
        }
        __syncthreads();
        if (tid < HEADS * SEQ) {
            int h = tid / SEQ, tq = tid - h * SEQ;
            float scv[SEQ]; float mx = -3.4e38f;
            for (int tk = 0; tk < SEQ; ++tk) {
                float d = 0.f;
                for (int dd = 0; dd < HDIM; ++dd)
                    d += qm[tq][h * HDIM + dd] * km[tk][h * HDIM + dd];
                scv[tk] = d * sc_att;
                mx = fmaxf(mx, scv[tk]);
            }
            float se = 0.f;
            for (int tk = 0; tk < SEQ; ++tk) { scv[tk] = __expf(scv[tk] - mx); se += scv[tk]; }
            float inv = 1.f / se;
            for (int dd = 0; dd < HDIM; ++dd) {
                float o = 0.f;
                for (int tk = 0; tk < SEQ; ++tk) o += scv[tk] * vm[tk][h * HDIM + dd];
                ao[tq][h * HDIM + dd] = o * inv;
            }
        }
        __syncthreads();
        for (int i = tid; i < SEQ * EMB; i += 256) {
            int t = i / EMB, e = i - t * EMB;
            const float* W = wo + (size_t)L * EMB * EMB + e * EMB;
            float a = bo[L * EMB + e];
            for (int j = 0; j < EMB; ++j) a += W[j] * ao[t][j];
            s[t][e] += a;
        }
        __syncthreads();
        if (tid < SEQ) {
            float m = 0.f;
            for (int e = 0; e < EMB; ++e) m += s[tid][e];
            m *= (1.f / EMB);
            float v = 0.f;
            for (int e = 0; e < EMB; ++e) { float d = s[tid][e] - m; v += d * d; }
            mu[tid] = m; rsd[tid] = rsqrtf(v * (1.f / EMB) + 1e-5f);
        }
        __syncthreads();
        for (int i = tid; i < SEQ * EMB; i += 256) {
            int t = i / EMB, e = i - t * EMB;
            tn[t][e] = (s[t][e] - mu[t]) * rsd[t] * ln2g[L * EMB + e] + ln2b[L * EMB + e];
        }
        __syncthreads();
        for (int i = tid; i < SEQ * 4 * EMB; i += 256) {
            int t = i / (4 * EMB), e = i - t * 4 * EMB;
            const float* W = fw1 + (size_t)L * 4 * EMB * EMB + e * EMB;
            float a = fb1[L * 4 * EMB + e];
            for (int j = 0; j < EMB; ++j) a += W[j] * tn[t][j];
            hid[t][e] = 0.5f * a * (1.f + erff(a * 0.70710678118f));   // exact GELU
        }
        __syncthreads();
        for (int i = tid; i < SEQ * EMB; i += 256) {
            int t = i / EMB, e = i - t * EMB;
            const float* W = fw2 + (size_t)L * EMB * 4 * EMB + e * 4 * EMB;
            float a = fb2[L * EMB + e];
            for (int j = 0; j < 4 * EMB; ++j) a += W[j] * hid[t][j];
            s[t][e] += a;
        }
        __syncthreads();
    }
    for (int i = tid; i < SEQ * EMB; i += 256) {
        int t = i / EMB, e = i - t * EMB;
        flat[(size_t)b * FEAT + i] = s[t][e];
    }
}

// ---------------- Kernel E: final FC via f32 WMMA (M=256,N=512,K=280) ----------------
// grid (8, 16), block 128 (4 waves; wave w -> N tile)
__global__ __launch_bounds__(128) void fc_wmma_kernel(
    const float* __restrict__ flat,  // (256,280)
    const float* __restrict__ fcw,   // (512,280)
    const float* __restrict__ fcb,   // (512)
    float* __restrict__ out)         // (256,512)
{
    __shared__ float As[16 * AS_STR];
    const int b0  = blockIdx.y * 16;
    const int tid = threadIdx.x;
    {
        // two-phase float4 staging: 16 rows x 70 vec4
        float4 v[9];
        #pragma unroll
        for (int j = 0; j < 9; ++j) {
            int i = tid + j * 128;
            if (i < 16 * (FC_K / 4)) {
                int r = i / (FC_K / 4), c = i - r * (FC_K / 4);
                v[j] = *(const float4*)(flat + (size_t)(b0 + r) * FC_K + c * 4);
            }
        }
        #pragma unroll
        for (int j = 0; j < 9; ++j) {
            int i = tid + j * 128;
            if (i < 16 * (FC_K / 4)) {
                int r = i / (FC_K / 4), c = i - r * (FC_K / 4);
                *(float4*)&As[r * AS_STR + c * 4] = v[j];
            }
        }
    }
    __syncthreads();

    const int lane  = tid & 31;
    const int wave  = tid >> 5;
    const int o0    = (blockIdx.x * 4 + wave) * 16;
    const int m     = lane & 15;
    const int khalf = (lane >> 4) << 1;

    v8f acc = {};
    #pragma unroll 5
    for (int k0 = 0; k0 < FC_K; k0 += 4) {
        int kk0 = k0 + khalf;
        v2f A = { As[m * AS_STR + kk0], As[m * AS_STR + kk0 + 1] };
        const float* wrow = fcw + (size_t)(o0 + m) * FC_K + kk0;
        v2f B = { wrow[0], wrow[1] };
        acc = __builtin_amdgcn_wmma_f32_16x16x4_f32(false, A, false, B,
                                                    (short)0, acc, false, false);
    }

    const int o     = o0 + m;
    const int rbase = (lane >> 4) * 8;
    const float bia = fcb[o];
    #pragma unroll
    for (int r = 0; r < 8; ++r)
        out[(size_t)(b0 + rbase + r) * OUTN + o] = acc[r] + bia;
}

// ---------------- host launch ----------------
extern "C" void kernel_launch(void* const* d_in, const int* in_sizes, int n_in,
                              void* d_out, int out_size, void* d_ws, size_t ws_size,
                              hipStream_t stream) {
    (void)in_sizes; (void)n_in; (void)out_size; (void)ws_size;
    const float* x      = (const float*)d_in[0];
    const float* conv1w = (const float*)d_in[1];
    const float* conv1b = (const float*)d_in[2];
    const float* conv2w = (const float*)d_in[3];
    const float* conv2b = (const float*)d_in[4];
    const float* bng    = (const float*)d_in[5];
    const float* bnb    = (const float*)d_in[6];
    const float* bnm    = (const float*)d_in[7];
    const float* bnv    = (const float*)d_in[8];
    const float* projw  = (const float*)d_in[9];
    const float* projb  = (const float*)d_in[10];
    const float* ln1g   = (const float*)d_in[11];
    const float* ln1b   = (const float*)d_in[12];
    const float* wq     = (const float*)d_in[13];
    const float* bq     = (const float*)d_in[14];
    const float* wk     = (const float*)d_in[15];
    const float* bk     = (const float*)d_in[16];
    const float* wv     = (const float*)d_in[17];
    const float* bv     = (const float*)d_in[18];
    const float* wo     = (const float*)d_in[19];
    const float* bo     = (const float*)d_in[20];
    const float* ln2g   = (const float*)d_in[21];
    const float* ln2b   = (const float*)d_in[22];
    const float* fw1    = (const float*)d_in[23];
    const float* fb1    = (const float*)d_in[24];
    const float* fw2    = (const float*)d_in[25];
    const float* fb2    = (const float*)d_in[26];
    const float* fcw    = (const float*)d_in[27];
    const float* fcb    = (const float*)d_in[28];

    char* ws = (char*)d_ws;
    float* Wc2    = (float*)(ws + 0);                               // 1600*48*4 = 307200 B
    float* biasC  = (float*)(ws + 307200);                          // 160 B
    float* eluOut = (float*)(ws + 307456);                          // 256*40*176*4 = 7208960 B
    float* pooled = (float*)(ws + 307456 + 7208960);                // 286720 B
    float* flat   = (float*)(ws + 307456 + 7208960 + 286720);       // 286720 B
    float* out    = (float*)d_out;

    fold_weights<<<300, 256, 0, stream>>>(conv1w, conv2w, bng, bnv, Wc2);
    fold_bias<<<1, 64, 0, stream>>>(conv1b, conv2w, conv2b, bng, bnb, bnm, bnv, biasC);
    dim3 gc(6, 256);
    conv_wmma_kernel<<<gc, 96, 0, stream>>>(x, Wc2, biasC, eluOut);
    pool_kernel<<<280, 256, 0, stream>>>(eluOut, pooled);
    transformer_kernel<<<256, 256, 0, stream>>>(pooled, projw, projb, ln1g, ln1b,
        wq, bq, wk, bk, wv, bv, wo, bo, ln2g, ln2b, fw1, fb1, fw2, fb2, flat);
    dim3 gf(8, 16);
    fc_wmma_kernel<<<gf, 128, 0, stream>>>(flat, fcw, fcb, out);
}